// GATLayer_13469017441031
// MI455X (gfx1250) — compile-verified
//
#include <hip/hip_runtime.h>

#define N_NODES   100000
#define N_EDGES   800000
#define IN_DIM    256
#define OUT_DIM   64
#define HEADS     4
#define FEATS     (HEADS * OUT_DIM)   // 256
#define NEG_SLOPE 0.2f
#define M_TILES   (N_NODES / 16)      // 6250

typedef float v2f __attribute__((ext_vector_type(2)));
typedef float v8f __attribute__((ext_vector_type(8)));

// ---------------------------------------------------------------------------
// Kernel 1: feat = h @ W via V_WMMA_F32_16X16X4_F32, one 16x64 strip (one
// head) per wave; epilogue fuses el/er attention dot products.
// ---------------------------------------------------------------------------
__global__ __launch_bounds__(128)
void gat_gemm_kernel(const float* __restrict__ hin,
                     const float* __restrict__ Wm,
                     const float* __restrict__ attn_l,
                     const float* __restrict__ attn_r,
                     float* __restrict__ feat,
                     float* __restrict__ el,
                     float* __restrict__ er)
{
    const int lane = threadIdx.x & 31;
    const int wave = threadIdx.x >> 5;
    const int mt   = blockIdx.x * 4 + wave;
    if (mt >= M_TILES) return;                 // uniform per-wave exit
    const int head = blockIdx.y;               // 0..3 -> 64-col strip
    const int m0   = mt * 16;
    const int n0   = head * OUT_DIM;
    const int l15  = lane & 15;
    const int lhi  = lane >> 4;                // 0 or 1
    const int row  = m0 + l15;
    const int koff = lhi * 2;

    v8f acc0 = {}, acc1 = {}, acc2 = {}, acc3 = {};
    const float* __restrict__ arow = hin + (size_t)row * IN_DIM;

    for (int k0 = 0; k0 < IN_DIM; k0 += 4) {
        const int kk = k0 + koff;
        v2f a = *(const v2f*)(arow + kk);
        const float* wr0 = Wm + (size_t)kk * FEATS + n0 + l15;
        const float* wr1 = wr0 + FEATS;
        v2f b0, b1, b2, b3;
        b0.x = wr0[0];   b0.y = wr1[0];
        b1.x = wr0[16];  b1.y = wr1[16];
        b2.x = wr0[32];  b2.y = wr1[32];
        b3.x = wr0[48];  b3.y = wr1[48];
        acc0 = __builtin_amdgcn_wmma_f32_16x16x4_f32(false, a, false, b0, (short)0, acc0, false, false);
        acc1 = __builtin_amdgcn_wmma_f32_16x16x4_f32(false, a, false, b1, (short)0, acc1, false, false);
        acc2 = __builtin_amdgcn_wmma_f32_16x16x4_f32(false, a, false, b2, (short)0, acc2, false, false);
        acc3 = __builtin_amdgcn_wmma_f32_16x16x4_f32(false, a, false, b3, (short)0, acc3, false, false);
    }

    // Attention weights for this head (lane l15 owns cols l15 + 16t)
    const float* al = attn_l + n0;
    const float* ar = attn_r + n0;
    const float al0 = al[l15], al1 = al[l15 + 16], al2 = al[l15 + 32], al3 = al[l15 + 48];
    const float ar0 = ar[l15], ar1 = ar[l15 + 16], ar2 = ar[l15 + 32], ar3 = ar[l15 + 48];

    #pragma unroll
    for (int r = 0; r < 8; ++r) {
        const int grow = m0 + r + 8 * lhi;     // C layout: VGPR r -> M=r (+8 for hi half)
        const float c0 = acc0[r], c1 = acc1[r], c2 = acc2[r], c3 = acc3[r];
        float* frow = feat + (size_t)grow * FEATS + n0 + l15;
        frow[0]  = c0;
        frow[16] = c1;
        frow[32] = c2;
        frow[48] = c3;
        float pl = c0 * al0 + c1 * al1 + c2 * al2 + c3 * al3;
        float pr = c0 * ar0 + c1 * ar1 + c2 * ar2 + c3 * ar3;
        // reduce over the 16 lanes holding this row (xor bits 0..3 stay in half)
        #pragma unroll
        for (int m = 1; m <= 8; m <<= 1) {
            pl += __shfl_xor(pl, m, 32);
            pr += __shfl_xor(pr, m, 32);
        }
        if (l15 == 0) {
            el[(size_t)grow * HEADS + head] = pl;
            er[(size_t)grow * HEADS + head] = pr;
        }
    }
}

// ---------------------------------------------------------------------------
// Kernel 2: init accumulators (ws is poisoned, must zero each call)
// ---------------------------------------------------------------------------
__global__ __launch_bounds__(256)
void gat_init_kernel(float* __restrict__ outacc,
                     float* __restrict__ mbuf,
                     float* __restrict__ denom)
{
    const size_t tid = (size_t)blockIdx.x * 256 + threadIdx.x;
    if (tid < (size_t)N_NODES * FEATS) outacc[tid] = 0.0f;
    if (tid < (size_t)N_NODES * HEADS) {
        mbuf[tid]  = -__builtin_inff();
        denom[tid] = 0.0f;
    }
}

__device__ __forceinline__ void atomicMaxFloat(float* addr, float val)
{
    // sign-split trick: correct total order for IEEE floats incl. -inf init
    if (val >= 0.0f) atomicMax((int*)addr, __float_as_int(val));
    else             atomicMin((unsigned int*)addr, __float_as_uint(val));
}

// ---------------------------------------------------------------------------
// Kernel 3: edge logits (leaky relu) + segment max over dst
// ---------------------------------------------------------------------------
__global__ __launch_bounds__(256)
void gat_logits_kernel(const float* __restrict__ el,
                       const float* __restrict__ er,
                       const int* __restrict__ src,
                       const int* __restrict__ dst,
                       float* __restrict__ ex,
                       float* __restrict__ mbuf)
{
    const int tid = blockIdx.x * 256 + threadIdx.x;
    if (tid >= N_EDGES * HEADS) return;
    const int e = tid >> 2, hd = tid & 3;
    const int s = src[e], d = dst[e];
    float v = el[s * HEADS + hd] + er[d * HEADS + hd];
    v = (v > 0.0f) ? v : NEG_SLOPE * v;
    ex[tid] = v;
    atomicMaxFloat(mbuf + (size_t)d * HEADS + hd, v);
}

// ---------------------------------------------------------------------------
// Kernel 4: ex = exp(e - m[dst]); denom[dst] += ex
// ---------------------------------------------------------------------------
__global__ __launch_bounds__(256)
void gat_expsum_kernel(const int* __restrict__ dst,
                       const float* __restrict__ mbuf,
                       float* __restrict__ ex,
                       float* __restrict__ denom)
{
    const int tid = blockIdx.x * 256 + threadIdx.x;
    if (tid >= N_EDGES * HEADS) return;
    const int e = tid >> 2, hd = tid & 3;
    const int d = dst[e];
    const float v = __expf(ex[tid] - mbuf[(size_t)d * HEADS + hd]);
    ex[tid] = v;
    atomicAdd(denom + (size_t)d * HEADS + hd, v);
}

// ---------------------------------------------------------------------------
// Kernel 5: out[dst] += (ex/denom[dst]) * feat[src]  (one edge per block,
// 256 threads = full H*D feature row, coalesced gather + float atomics)
// ---------------------------------------------------------------------------
__global__ __launch_bounds__(256)
void gat_aggregate_kernel(const float* __restrict__ feat,
                          const float* __restrict__ ex,
                          const float* __restrict__ denom,
                          const int* __restrict__ src,
                          const int* __restrict__ dst,
                          float* __restrict__ outacc)
{
    const int e  = blockIdx.x;
    const int c  = threadIdx.x;        // 0..255
    const int hd = c >> 6;
    const int s  = src[e], d = dst[e];
    const float a = ex[(size_t)e * HEADS + hd] / denom[(size_t)d * HEADS + hd];
    atomicAdd(outacc + (size_t)d * FEATS + c, a * feat[(size_t)s * FEATS + c]);
}

// ---------------------------------------------------------------------------
// Kernel 6: residual + bias, mean over heads
// ---------------------------------------------------------------------------
__global__ __launch_bounds__(256)
void gat_finalize_kernel(const float* __restrict__ outacc,
                         const float* __restrict__ hin,
                         const float* __restrict__ bias,
                         float* __restrict__ out)
{
    const int tid = blockIdx.x * 256 + threadIdx.x;
    if (tid >= N_NODES * OUT_DIM) return;
    const int n = tid >> 6, dd = tid & 63;
    float s = 0.0f;
    #pragma unroll
    for (int hd = 0; hd < HEADS; ++hd) {
        const int c = hd * OUT_DIM + dd;
        s += outacc[(size_t)n * FEATS + c] + hin[(size_t)n * FEATS + c] + bias[c];
    }
    out[tid] = s * 0.25f;
}

// ---------------------------------------------------------------------------
extern "C" void kernel_launch(void* const* d_in, const int* in_sizes, int n_in,
                              void* d_out, int out_size, void* d_ws, size_t ws_size,
                              hipStream_t stream)
{
    const float* hin    = (const float*)d_in[0];
    const float* Wm     = (const float*)d_in[1];
    const float* attn_l = (const float*)d_in[2];
    const float* attn_r = (const float*)d_in[3];
    const float* bias   = (const float*)d_in[4];
    const int*   src    = (const int*)d_in[5];
    const int*   dst    = (const int*)d_in[6];
    float*       out    = (float*)d_out;

    char* ws = (char*)d_ws;
    size_t off = 0;
    auto carve = [&](size_t bytes) -> float* {
        float* p = (float*)(ws + off);
        off += (bytes + 255) & ~(size_t)255;
        return p;
    };
    float* feat   = carve((size_t)N_NODES * FEATS * sizeof(float));  // 102.4 MB
    float* outacc = carve((size_t)N_NODES * FEATS * sizeof(float));  // 102.4 MB
    float* el     = carve((size_t)N_NODES * HEADS * sizeof(float));
    float* er     = carve((size_t)N_NODES * HEADS * sizeof(float));
    float* mbuf   = carve((size_t)N_NODES * HEADS * sizeof(float));
    float* denom  = carve((size_t)N_NODES * HEADS * sizeof(float));
    float* ex     = carve((size_t)N_EDGES * HEADS * sizeof(float));  // 12.8 MB

    // 1) projection + attention halves (WMMA)
    dim3 ggrid((M_TILES + 3) / 4, HEADS);
    gat_gemm_kernel<<<ggrid, 128, 0, stream>>>(hin, Wm, attn_l, attn_r, feat, el, er);

    // 2) zero / -inf init of accumulators
    gat_init_kernel<<<(N_NODES * FEATS + 255) / 256, 256, 0, stream>>>(outacc, mbuf, denom);

    // 3) logits + segment max
    gat_logits_kernel<<<(N_EDGES * HEADS + 255) / 256, 256, 0, stream>>>(el, er, src, dst, ex, mbuf);

    // 4) exp + segment sum
    gat_expsum_kernel<<<(N_EDGES * HEADS + 255) / 256, 256, 0, stream>>>(dst, mbuf, ex, denom);

    // 5) normalized weighted aggregation (scatter)
    gat_aggregate_kernel<<<N_EDGES, 256, 0, stream>>>(feat, ex, denom, src, dst, outacc);

    // 6) residual + bias + head mean
    gat_finalize_kernel<<<(N_NODES * OUT_DIM + 255) / 256, 256, 0, stream>>>(outacc, hin, bias, out);
}